// SPDBatchNorm_65189013619292
// MI455X (gfx1250) — compile-verified
//
#include <hip/hip_runtime.h>
#include <hip/hip_bf16.h>

// SPD batch-norm for MI455X (gfx1250): all per-element matrix functions
// (logm / powm) implemented as 32x32 matmul chains on V_WMMA_F32_16X16X4_F32.
// One wave32 per batch element; matrices staged in LDS (stride 34 floats:
// conflict-free across 64 banks, 8B aligned for b64 A-fragment loads).

#define NB        65536
#define MSTR      34
#define MELTS     (32 * MSTR)
#define WAVES     2
#define BLOCK     (WAVES * 32)
#define HEAVY_GRID 2048          // 2048 blocks * 2 waves = 4096 slots, 16 elems each
#define EPS_BN    1e-5f

typedef float v2f __attribute__((ext_vector_type(2)));
typedef float v8f __attribute__((ext_vector_type(8)));

// ------------------------------------------------------------------
// Wave-level 32x32 f32 matmul: C = A * B, all operands in LDS.
// Uses V_WMMA_F32_16X16X4_F32 (A 16x4, B 4x16, C 16x16), 2x2 tiles x 8 k-steps.
// Fragment layouts per CDNA5 ISA 7.12.2:
//   A 16x4 : lanes 0-15 -> K={k0,k0+1}, lanes 16-31 -> K={k0+2,k0+3}
//   B 4x16 : VGPR v, lane half h -> row K = k0 + 2h + v, col N = lane&15
//   C 16x16: VGPR r, lane half h -> row M = r + 8h, col N = lane&15
// ------------------------------------------------------------------
__device__ __forceinline__ void wave_mm32(const float* A, const float* Bm,
                                          float* C, int lane) {
  const int lh = lane >> 4;
  const int l  = lane & 15;
#pragma unroll
  for (int ti = 0; ti < 2; ++ti) {
#pragma unroll
    for (int tj = 0; tj < 2; ++tj) {
      v8f acc = {0.f, 0.f, 0.f, 0.f, 0.f, 0.f, 0.f, 0.f};
#pragma unroll
      for (int k0 = 0; k0 < 32; k0 += 4) {
        const float* ap = A + (ti * 16 + l) * MSTR + (k0 + lh * 2);
        v2f av; av.x = ap[0]; av.y = ap[1];
        const float* bp = Bm + (k0 + lh * 2) * MSTR + tj * 16 + l;
        v2f bv; bv.x = bp[0]; bv.y = bp[MSTR];
        acc = __builtin_amdgcn_wmma_f32_16x16x4_f32(
            false, av, false, bv, (short)0, acc, false, false);
      }
      float* cp = C + (ti * 16 + lh * 8) * MSTR + tj * 16 + l;
#pragma unroll
      for (int r = 0; r < 8; ++r) cp[r * MSTR] = acc[r];
    }
  }
}

// ---------------- wave-level elementwise helpers (LDS) ----------------
__device__ __forceinline__ void wave_scale_addI(float* M, float s, float d, int lane) {
  for (int i = lane; i < 1024; i += 32) {
    int r = i >> 5, c = i & 31;
    float v = M[r * MSTR + c] * s;
    if (r == c) v += d;
    M[r * MSTR + c] = v;
  }
}
__device__ __forceinline__ void wave_copy_scale_addI(float* D, const float* S,
                                                     float s, float d, int lane) {
  for (int i = lane; i < 1024; i += 32) {
    int r = i >> 5, c = i & 31;
    float v = S[r * MSTR + c] * s;
    if (r == c) v += d;
    D[r * MSTR + c] = v;
  }
}
__device__ __forceinline__ void wave_set_scaledI(float* M, float d, int lane) {
  for (int i = lane; i < 1024; i += 32) {
    int r = i >> 5, c = i & 31;
    M[r * MSTR + c] = (r == c) ? d : 0.f;
  }
}
__device__ __forceinline__ void wave_sym(float* M, int lane) {
  for (int i = lane; i < 1024; i += 32) {
    int r = i >> 5, c = i & 31;
    if (r < c) {
      float v = 0.5f * (M[r * MSTR + c] + M[c * MSTR + r]);
      M[r * MSTR + c] = v;
      M[c * MSTR + r] = v;
    }
  }
}
// Gershgorin upper bound on lambda_max: max row abs-sum (lane = row).
__device__ __forceinline__ float wave_gersh(const float* M, int lane) {
  float s = 0.f;
  for (int c = 0; c < 32; ++c) s += fabsf(M[lane * MSTR + c]);
#pragma unroll
  for (int m = 16; m >= 1; m >>= 1) s = fmaxf(s, __shfl_xor(s, m, 32));
  return s;
}

// ------------------------------------------------------------------
// L = log(M), M SPD in buffer X. Inverse scaling-and-squaring:
//   c = Gershgorin bound -> eigenvalues of M/c in (0,1]
//   4 coupled Newton-Schulz sqrts -> R = (M/c)^(1/16), eig in [~0.78,1]
//   log(I+E) via degree-8 Horner series, L = 16*log(R) + log(c)*I
// Result is ALWAYS left in buffer T; X,Z,U become scratch.
// ------------------------------------------------------------------
__device__ __forceinline__ void wave_logm(float* X, float* Z, float* T, float* U,
                                          int lane) {
  float c = wave_gersh(X, lane);
  wave_scale_addI(X, 1.0f / c, 0.f, lane);
  float* Y  = X;
  float* Zb = Z;
  float* Tb = T;
  float* Ub = U;
  const int iters[4] = {9, 7, 6, 5};
#pragma clang loop unroll(disable)
  for (int s = 0; s < 4; ++s) {
    wave_set_scaledI(Zb, 1.f, lane);
#pragma clang loop unroll(disable)
    for (int it = 0; it < iters[s]; ++it) {
      wave_mm32(Zb, Y, Tb, lane);               // T = Z*Y
      wave_scale_addI(Tb, -0.5f, 1.5f, lane);   // T = 1.5I - 0.5 Z*Y
      wave_mm32(Y, Tb, Ub, lane);               // newY = Y*T
      wave_mm32(Tb, Zb, Y, lane);               // newZ = T*Z (into old Y)
      float* ny = Ub; float* nz = Y;
      Ub = Zb; Y = ny; Zb = nz;                 // rotate {Y,Z,U}; T fixed
    }
  }
  wave_scale_addI(Y, 1.f, -1.f, lane);          // E = R - I
  wave_set_scaledI(Zb, -1.0f / 8.0f, lane);     // P = c8*I
#pragma clang loop unroll(disable)
  for (int j = 7; j >= 1; --j) {
    wave_mm32(Y, Zb, Tb, lane);                 // T = E*P
    float cj = ((j & 1) ? 1.0f : -1.0f) / (float)j;
    wave_copy_scale_addI(Zb, Tb, 1.f, cj, lane);
  }
  wave_mm32(Y, Zb, Tb, lane);                   // T = E*P = log(I+E)
  wave_scale_addI(Tb, 16.f, logf(c), lane);     // L = 16*logR + log(c)I  (in T)
}

// exp(A): scaling (1/32) + degree-8 Taylor Horner + 5 squarings.
// A destroyed; result in P; Tb scratch.
__device__ __forceinline__ void wave_expm(float* A, float* P, float* Tb, int lane) {
  wave_scale_addI(A, 1.0f / 32.0f, 0.f, lane);
  wave_set_scaledI(P, 1.f, lane);
#pragma clang loop unroll(disable)
  for (int j = 8; j >= 1; --j) {
    wave_mm32(A, P, Tb, lane);
    wave_copy_scale_addI(P, Tb, 1.0f / (float)j, 1.0f, lane);
  }
#pragma clang loop unroll(disable)
  for (int s = 0; s < 5; ++s) {
    wave_mm32(P, P, Tb, lane);
    wave_copy_scale_addI(P, Tb, 1.f, 0.f, lane);
  }
}

// load x[e], whiten with Sg (LDS-staged inv-sqrt), symmetrize, logm.  L -> T.
__device__ __forceinline__ void load_whiten_logm(const float* xg, const float* Sg,
                                                 float* X, float* Z, float* T,
                                                 float* U, int lane) {
#pragma clang loop unroll(disable)
  for (int r = 0; r < 32; ++r) Z[r * MSTR + lane] = xg[r * 32 + lane];
  wave_mm32(Sg, Z, T, lane);
  wave_mm32(T, Sg, X, lane);
  wave_sym(X, lane);
  wave_logm(X, Z, T, U, lane);
}

// ---------------- block-level (single 32x32 matrix) helpers ----------------
__device__ __forceinline__ float blk_dot(const float* A, const float* Bm, int i, int j) {
  float a = 0.f;
  for (int k = 0; k < 32; ++k) a += A[i * 32 + k] * Bm[k * 32 + j];
  return a;
}
// Coupled Newton-Schulz: Y -> sqrt(A), Z -> A^{-1/2}.  Destroys A.
__device__ void blk_ns(float* A, float* Y, float* Z, float* rs, float* cSh,
                       int t, int i, int j, int iters) {
  if (t < 32) {
    float s = 0.f;
    for (int k = 0; k < 32; ++k) s += fabsf(A[t * 32 + k]);
    rs[t] = s;
  }
  __syncthreads();
  if (t == 0) {
    float mx = 0.f;
    for (int k = 0; k < 32; ++k) mx = fmaxf(mx, rs[k]);
    *cSh = mx;
  }
  __syncthreads();
  float c = *cSh;
  Y[t] = A[t] / c;
  Z[t] = (i == j) ? 1.f : 0.f;
  __syncthreads();
  for (int it = 0; it < iters; ++it) {
    float zy = blk_dot(Z, Y, i, j);
    __syncthreads();
    A[t] = ((i == j) ? 1.5f : 0.f) - 0.5f * zy;   // T (reuse A)
    __syncthreads();
    float ny = blk_dot(Y, A, i, j);
    float nz = blk_dot(A, Z, i, j);
    __syncthreads();
    Y[t] = ny; Z[t] = nz;
    __syncthreads();
  }
  float sc = sqrtf(c);
  Y[t] *= sc;
  Z[t] /= sc;
  __syncthreads();
}

// ------------------------------- kernels -------------------------------
__global__ void k_zero(float* p, int n) {
  int i = blockIdx.x * blockDim.x + threadIdx.x;
  if (i < n) p[i] = 0.f;
}

__global__ __launch_bounds__(256) void k_mean_acc(const float* __restrict__ x,
                                                  float* __restrict__ acc) {
  int t = threadIdx.x;
  int b0 = blockIdx.x * 64;
  float s0 = 0.f, s1 = 0.f, s2 = 0.f, s3 = 0.f;
  for (int b = b0; b < b0 + 64; ++b) {
    const float* xb = x + (size_t)b * 1024;
    s0 += xb[t]; s1 += xb[t + 256]; s2 += xb[t + 512]; s3 += xb[t + 768];
  }
  atomicAdd(&acc[t], s0);
  atomicAdd(&acc[t + 256], s1);
  atomicAdd(&acc[t + 512], s2);
  atomicAdd(&acc[t + 768], s3);
}

__global__ __launch_bounds__(1024) void k_prep1(const float* meanAcc,
                                                float* inv_s0, float* s0) {
  __shared__ float A[1024], Y[1024], Z[1024], rs[32];
  __shared__ float cSh;
  int t = threadIdx.x, i = t >> 5, j = t & 31;
  A[t] = meanAcc[t] * (1.0f / 65536.0f);
  __syncthreads();
  blk_ns(A, Y, Z, rs, &cSh, t, i, j, 14);
  s0[t] = Y[t];
  inv_s0[t] = Z[t];
}

__global__ __launch_bounds__(BLOCK) void k_logacc(const float* __restrict__ x,
                                                  const float* __restrict__ invs0,
                                                  float* __restrict__ logAcc) {
  __shared__ float sStage[MELTS];
  __shared__ float sAcc[1024];
  __shared__ float sBuf[WAVES * 4 * MELTS];
  int tid = threadIdx.x, lane = tid & 31, w = tid >> 5;
  for (int i = tid; i < 1024; i += BLOCK) {
    sStage[(i >> 5) * MSTR + (i & 31)] = invs0[i];
    sAcc[i] = 0.f;
  }
  __syncthreads();
  float* X = sBuf + w * 4 * MELTS;
  float* Z = X + MELTS;
  float* T = Z + MELTS;
  float* U = T + MELTS;
  int slot = blockIdx.x * WAVES + w, ns = gridDim.x * WAVES;
  for (int e = slot; e < NB; e += ns) {
    load_whiten_logm(x + (size_t)e * 1024, sStage, X, Z, T, U, lane);
    for (int i = lane; i < 1024; i += 32)
      atomicAdd(&sAcc[i], T[(i >> 5) * MSTR + (i & 31)]);
  }
  __syncthreads();
  for (int i = tid; i < 1024; i += BLOCK) atomicAdd(&logAcc[i], sAcc[i]);
}

__global__ __launch_bounds__(1024) void k_prep2(const float* logAcc, const float* s0g,
                                                const float* shift, float* inv_s,
                                                float* s_sqrt) {
  __shared__ float A[1024], Q[1024], C[1024], Y[1024], Z[1024], rs[32];
  __shared__ float cSh;
  int t = threadIdx.x, i = t >> 5, j = t & 31;
  A[t] = logAcc[t] * (1.0f / 65536.0f) * (1.0f / 16.0f);  // Lavg / 2^4
  C[t] = s0g[t];
  Q[t] = (i == j) ? 1.f : 0.f;
  __syncthreads();
  for (int jj = 8; jj >= 1; --jj) {                        // exp Taylor Horner
    float m = blk_dot(A, Q, i, j);
    __syncthreads();
    Q[t] = ((i == j) ? 1.f : 0.f) + m / (float)jj;
    __syncthreads();
  }
  for (int s = 0; s < 4; ++s) {                            // 4 squarings
    float m = blk_dot(Q, Q, i, j);
    __syncthreads();
    Q[t] = m;
    __syncthreads();
  }
  float m1 = blk_dot(C, Q, i, j);                          // mean = s0*exp(L)*s0
  __syncthreads();
  A[t] = m1;
  __syncthreads();
  float m2 = blk_dot(A, C, i, j);
  __syncthreads();
  A[t] = m2;
  __syncthreads();
  float sv = 0.5f * (A[i * 32 + j] + A[j * 32 + i]);       // sym
  __syncthreads();
  A[t] = sv;
  __syncthreads();
  blk_ns(A, Y, Z, rs, &cSh, t, i, j, 14);                  // inv_s = mean^{-1/2}
  inv_s[t] = Z[t];
  A[t] = shift[t];
  __syncthreads();
  blk_ns(A, Y, Z, rs, &cSh, t, i, j, 14);                  // s_sqrt = shift^{1/2}
  s_sqrt[t] = Y[t];
}

__global__ __launch_bounds__(BLOCK) void k_var(const float* __restrict__ x,
                                               const float* __restrict__ invs,
                                               float* __restrict__ varAcc) {
  __shared__ float sStage[MELTS];
  __shared__ float sBuf[WAVES * 4 * MELTS];
  __shared__ float sVar;
  int tid = threadIdx.x, lane = tid & 31, w = tid >> 5;
  for (int i = tid; i < 1024; i += BLOCK)
    sStage[(i >> 5) * MSTR + (i & 31)] = invs[i];
  if (tid == 0) sVar = 0.f;
  __syncthreads();
  float* X = sBuf + w * 4 * MELTS;
  float* Z = X + MELTS;
  float* T = Z + MELTS;
  float* U = T + MELTS;
  int slot = blockIdx.x * WAVES + w, ns = gridDim.x * WAVES;
  float part = 0.f;
  for (int e = slot; e < NB; e += ns) {
    load_whiten_logm(x + (size_t)e * 1024, sStage, X, Z, T, U, lane);
    for (int i = lane; i < 1024; i += 32) {
      float v = T[(i >> 5) * MSTR + (i & 31)];
      part += v * v;
    }
  }
#pragma unroll
  for (int m = 16; m >= 1; m >>= 1) part += __shfl_xor(part, m, 32);
  if (lane == 0) atomicAdd(&sVar, part);
  __syncthreads();
  if (tid == 0) atomicAdd(varAcc, sVar);
}

__global__ void k_p(const float* varAcc, const float* scale, float* pOut) {
  if (threadIdx.x == 0 && blockIdx.x == 0) {
    float v = varAcc[0] * (1.0f / 65536.0f);
    pOut[0] = scale[0] / (sqrtf(v) + EPS_BN);
  }
}

__global__ __launch_bounds__(BLOCK) void k_final(const float* __restrict__ x,
                                                 const float* __restrict__ invs,
                                                 const float* __restrict__ ssq,
                                                 const float* __restrict__ pPtr,
                                                 float* __restrict__ out) {
  __shared__ float sW[MELTS];
  __shared__ float sQ[MELTS];
  __shared__ float sBuf[WAVES * 4 * MELTS];
  int tid = threadIdx.x, lane = tid & 31, w = tid >> 5;
  for (int i = tid; i < 1024; i += BLOCK) {
    int r = i >> 5, c = i & 31;
    sW[r * MSTR + c] = invs[i];
    sQ[r * MSTR + c] = ssq[i];
  }
  __syncthreads();
  float p = pPtr[0];
  float* X = sBuf + w * 4 * MELTS;
  float* Z = X + MELTS;
  float* T = Z + MELTS;
  float* U = T + MELTS;
  int slot = blockIdx.x * WAVES + w, ns = gridDim.x * WAVES;
  for (int e = slot; e < NB; e += ns) {
    load_whiten_logm(x + (size_t)e * 1024, sW, X, Z, T, U, lane);  // L in T
    wave_scale_addI(T, p, 0.f, lane);                              // p*L
    wave_expm(T, X, Z, lane);                                      // xp in X
    wave_mm32(sQ, X, U, lane);                                     // s_sqrt*xp
    wave_mm32(U, sQ, Z, lane);                                     // *s_sqrt
    wave_sym(Z, lane);
    float* og = out + (size_t)e * 1024;
#pragma clang loop unroll(disable)
    for (int r = 0; r < 32; ++r) og[r * 32 + lane] = Z[r * MSTR + lane];
  }
}

// ------------------------------- launch -------------------------------
extern "C" void kernel_launch(void* const* d_in, const int* in_sizes, int n_in,
                              void* d_out, int out_size, void* d_ws, size_t ws_size,
                              hipStream_t stream) {
  (void)in_sizes; (void)n_in; (void)out_size; (void)ws_size;
  const float* x     = (const float*)d_in[0];
  const float* shift = (const float*)d_in[1];
  const float* scale = (const float*)d_in[2];
  float* out = (float*)d_out;
  float* ws  = (float*)d_ws;

  float* meanAcc = ws + 0;      // 1024
  float* logAcc  = ws + 1024;   // 1024
  float* varAcc  = ws + 2048;   // 1
  float* pScal   = ws + 2049;   // 1
  float* inv_s0  = ws + 3072;   // 1024
  float* s0      = ws + 4096;   // 1024
  float* inv_s   = ws + 5120;   // 1024
  float* s_sqrt  = ws + 6144;   // 1024

  k_zero<<<9, 256, 0, stream>>>(ws, 2050);
  k_mean_acc<<<1024, 256, 0, stream>>>(x, meanAcc);
  k_prep1<<<1, 1024, 0, stream>>>(meanAcc, inv_s0, s0);
  k_logacc<<<HEAVY_GRID, BLOCK, 0, stream>>>(x, inv_s0, logAcc);
  k_prep2<<<1, 1024, 0, stream>>>(logAcc, s0, shift, inv_s, s_sqrt);
  k_var<<<HEAVY_GRID, BLOCK, 0, stream>>>(x, inv_s, varAcc);
  k_p<<<1, 32, 0, stream>>>(varAcc, scale, pScal);
  k_final<<<HEAVY_GRID, BLOCK, 0, stream>>>(x, inv_s, s_sqrt, pScal, out);
}